// RoiAlignModel_22686017257694
// MI455X (gfx1250) — compile-verified
//
#include <hip/hip_runtime.h>

// RoIAlign (torchvision semantics, ALIGNED=true, SCALE=0.25, POOL=7, RATIO=2)
// reformulated as per-(roi,channel) separable GEMMs:
//   out_c(7x7) = Wy(7x28) * F_c(28x28) * Wx(28x7)
// executed with V_WMMA_F32_16X16X4_F32 (padded 16x32 / 32x448 tiles),
// batching 16 channels per WMMA tile. One workgroup per ROI (4 waves).
// Gather is double-buffered through LDS using GLOBAL_LOAD_ASYNC_TO_LDS_B32
// (ASYNCcnt), overlapping HBM gather latency with the WMMA stages.

#define POOL    7
#define NFC     256     // channels
#define FH      200
#define FW      304
#define NS      28      // 14 samples * 2 corners per axis
#define NSP     32      // padded slot count (WMMA K dim)
#define NCHUNK  16      // channels per chunk (WMMA tile width)
#define FELTS   (NSP * NS * NCHUNK)   // 14336 floats per F buffer

typedef __attribute__((ext_vector_type(2))) float v2f;
typedef __attribute__((ext_vector_type(8))) float v8f;

#if __has_builtin(__builtin_amdgcn_global_load_async_to_lds_b32)
#define HAS_ASYNC_LDS 1
typedef const __attribute__((address_space(1))) float* g_cf32p;  // step 1: AS cast
typedef __attribute__((address_space(1))) int*         g_i32p;   // step 2: type cast
typedef __attribute__((address_space(3))) float*       l_f32p;
typedef __attribute__((address_space(3))) int*         l_i32p;
#else
#define HAS_ASYNC_LDS 0
#endif

__device__ __forceinline__ void wait_async_done() {
#if HAS_ASYNC_LDS
#if __has_builtin(__builtin_amdgcn_s_wait_asynccnt)
  __builtin_amdgcn_s_wait_asynccnt(0);
#else
  asm volatile("s_wait_asynccnt 0x0" ::: "memory");
#endif
#endif
}

__device__ __forceinline__ v8f wmma4(v2f a, v2f b, v8f c) {
  // D(16x16,f32) = A(16x4,f32) * B(4x16,f32) + C
  return __builtin_amdgcn_wmma_f32_16x16x4_f32(
      /*neg_a=*/false, a, /*neg_b=*/false, b,
      /*c_mod=*/(short)0, c, /*reuse_a=*/false, /*reuse_b=*/false);
}

// Gather one 16-channel chunk of corner samples into an LDS F buffer.
// Layout: F[js(32)][is(28)][cl(16)], element = feats[b, cb+cl, Y[js], X[is]].
__device__ __forceinline__ void gather_chunk(const float* __restrict__ feats,
                                             float* __restrict__ Fdst,
                                             const int* __restrict__ Xs,
                                             const int* __restrict__ Ys,
                                             int base, int tid) {
  for (int idx = tid; idx < FELTS; idx += 128) {
    const int cl   = idx & 15;
    const int rest = idx >> 4;
    const int is   = rest % NS;
    const int js   = rest / NS;
    const int off  = base + cl * (FH * FW) + Ys[js] * FW + Xs[is];
#if HAS_ASYNC_LDS
    __builtin_amdgcn_global_load_async_to_lds_b32(
        (g_i32p)(g_cf32p)(feats + off), (l_i32p)(l_f32p)(Fdst + idx), 0, 0);
#else
    Fdst[idx] = feats[off];
#endif
  }
}

__global__ void __launch_bounds__(128)
roi_align_wmma(const float* __restrict__ feats, const float* __restrict__ rois,
               float* __restrict__ out, int K)
{
  extern __shared__ char smem_raw[];
  float* F0   = (float*)smem_raw;           // [32 js][28 is][16 cl] gather buf 0
  float* F1   = F0 + FELTS;                 // gather buf 1 (double buffer)
  float* V    = F1 + FELTS;                 // [16 py][32 is * 16 cl] stage-A out
  float* WyL  = V + 16 * 512;               // [16 m][32 k]
  float* WxL  = WyL + 512;                  // [32 k][16 n]
  float* wcol = WxL + 512;                  // [32]
  float* wrow = wcol + 32;                  // [32]
  int*   Xs   = (int*)(wrow + 32);          // [32]
  int*   Ys   = Xs + 32;                    // [32]

  const int k    = blockIdx.x;
  const int tid  = threadIdx.x;
  const int lane = tid & 31;
  const int wave = tid >> 5;
  const int half = lane >> 4;   // 0: lanes 0-15, 1: lanes 16-31
  const int l16  = lane & 15;

  // ---- per-ROI geometry ----
  const float* r = rois + (size_t)k * 5;
  const int   b  = (int)r[0];
  const float x1 = r[1] * 0.25f - 0.5f;
  const float y1 = r[2] * 0.25f - 0.5f;
  const float x2 = r[3] * 0.25f - 0.5f;
  const float y2 = r[4] * 0.25f - 0.5f;
  const float bin_w = (x2 - x1) * (1.0f / POOL);
  const float bin_h = (y2 - y1) * (1.0f / POOL);

  // slot s = 2*i + p : sample i (0..13), corner p (0:lo, 1:hi)
  if (tid < NSP) {
    if (tid < NS) {
      const int   i    = tid >> 1;
      const int   p    = tid & 1;
      const float frac = (float)(i >> 1) + ((float)(i & 1) + 0.5f) * 0.5f;
      {
        float coord = x1 + frac * bin_w;
        bool  valid = (coord >= -1.0f) && (coord <= (float)FW);
        float c  = fmaxf(coord, 0.0f);
        int   lo = min((int)floorf(c), FW - 1);
        int   hi = min(lo + 1, FW - 1);
        float l  = c - (float)lo;
        Xs[tid]   = p ? hi : lo;
        wcol[tid] = (p ? l : 1.0f - l) * (valid ? 0.5f : 0.0f); // 0.5 = x-half of /4 mean
      }
      {
        float coord = y1 + frac * bin_h;
        bool  valid = (coord >= -1.0f) && (coord <= (float)FH);
        float c  = fmaxf(coord, 0.0f);
        int   lo = min((int)floorf(c), FH - 1);
        int   hi = min(lo + 1, FH - 1);
        float l  = c - (float)lo;
        Ys[tid]   = p ? hi : lo;
        wrow[tid] = (p ? l : 1.0f - l) * (valid ? 0.5f : 0.0f);
      }
    } else {                       // padded slots 28..31: safe index, zero weight
      Xs[tid] = 0; Ys[tid] = 0; wcol[tid] = 0.0f; wrow[tid] = 0.0f;
    }
  }
  __syncthreads();

  // ---- build padded banded weight matrices, zero V pad columns ----
  for (int idx = tid; idx < 512; idx += 128) {          // Wy [16 py][32 js]
    int rr = idx >> 5, cc = idx & 31;
    WyL[idx] = (rr < POOL && cc < NS && (cc >> 2) == rr) ? wrow[cc] : 0.0f;
  }
  for (int idx = tid; idx < 512; idx += 128) {          // Wx [32 is][16 px]
    int rr = idx >> 4, cc = idx & 15;
    WxL[idx] = (rr < NS && (rr >> 2) == cc) ? wcol[rr] : 0.0f;
  }
  for (int idx = tid; idx < 16 * 64; idx += 128) {      // V cols 448..511 = 0 (pad K)
    int row = idx >> 6, col = idx & 63;
    V[row * 512 + 448 + col] = 0.0f;
  }
  __syncthreads();

  // ---- hoist ROI-constant WMMA fragments ----
  // A 16x4 f32 layout (ISA 7.12.2): lanes0-15 -> VGPR0=K0,VGPR1=K1; lanes16-31 -> K2,K3
  // B 4x16 f32 layout (row-striped, analogous to C/D): VGPR v: lanes0-15 K=v, lanes16-31 K=v+2
  v2f aWy[8], bWx[8];
#pragma unroll
  for (int kk = 0; kk < 8; ++kk) {
    const int k0 = 4 * kk + 2 * half;
    aWy[kk].x = WyL[l16 * 32 + k0];
    aWy[kk].y = WyL[l16 * 32 + k0 + 1];
    bWx[kk].x = WxL[k0 * 16 + l16];
    bWx[kk].y = WxL[(k0 + 1) * 16 + l16];
  }

  const int base0 = b * (NFC * FH * FW);

  // Prime the pipeline: gather chunk 0 into F0.
  gather_chunk(feats, F0, Xs, Ys, base0, tid);
  wait_async_done();
  __syncthreads();

  // ---- channel-chunk loop: (prefetch next) -> stage A (Wy*F) -> stage B (V*Wx) ----
  for (int ci = 0; ci < NFC / NCHUNK; ++ci) {
    const int cb = ci * NCHUNK;
    float* Fc = (ci & 1) ? F1 : F0;
    float* Fn = (ci & 1) ? F0 : F1;

    // Issue async gather for the next chunk; overlaps with WMMA stages below.
    if (ci + 1 < NFC / NCHUNK)
      gather_chunk(feats, Fn, Xs, Ys, base0 + (cb + NCHUNK) * (FH * FW), tid);

    // Stage A: V[16 py][448] = Wy(16x32) @ F(32x448); 7 N-tiles per wave.
    // Constant trip count -> no exec masking around WMMA (EXEC stays all-1s).
#pragma unroll
    for (int tt = 0; tt < 7; ++tt) {
      const int t = wave + 4 * tt;          // 0..27, wave-uniform
      v8f acc = {};
#pragma unroll
      for (int kk = 0; kk < 8; ++kk) {
        const int js0 = 4 * kk + 2 * half;
        v2f bb;
        bb.x = Fc[(js0 * NS + t) * NCHUNK + l16];
        bb.y = Fc[((js0 + 1) * NS + t) * NCHUNK + l16];
        acc = wmma4(aWy[kk], bb, acc);
      }
#pragma unroll
      for (int v = 0; v < 8; ++v)   // D: VGPR v -> row v (lanes0-15) / v+8 (lanes16-31)
        V[(v + 8 * half) * 512 + t * 16 + l16] = acc[v];
    }
    __syncthreads();

    // Stage B: per py tile, out[16 ch][16 px] = A(ch x is) @ Wx(is x px).
    // 2 tiles per wave; tile mt==7 (wave 3) is a dummy on zeroed V row 7 and
    // only its store is suppressed, so WMMA always runs with full EXEC.
#pragma unroll
    for (int tt = 0; tt < 2; ++tt) {
      const int mt = wave + 4 * tt;         // 0..7, wave-uniform
      v8f acc = {};
#pragma unroll
      for (int kk = 0; kk < 8; ++kk) {
        const int is0 = 4 * kk + 2 * half;
        v2f aa;                     // A[cl][is] = V[mt][is*16+cl]
        aa.x = V[mt * 512 + is0 * 16 + l16];
        aa.y = V[mt * 512 + (is0 + 1) * 16 + l16];
        acc = wmma4(aa, bWx[kk], acc);
      }
      if (mt < POOL && l16 < POOL) {        // n = px, rows = channel
#pragma unroll
        for (int v = 0; v < 8; ++v) {
          const int cch = cb + v + 8 * half;
          out[(((size_t)k * NFC + cch) * POOL + mt) * POOL + l16] = acc[v];
        }
      }
    }

    // Next F buffer must be resident (per-wave ASYNCcnt), then block-wide sync
    // (also frees V for the next stage A and Fc for the gather after next).
    wait_async_done();
    __syncthreads();
  }
}

extern "C" void kernel_launch(void* const* d_in, const int* in_sizes, int n_in,
                              void* d_out, int out_size, void* d_ws, size_t ws_size,
                              hipStream_t stream) {
  (void)n_in; (void)out_size; (void)d_ws; (void)ws_size;
  const float* feats = (const float*)d_in[0];
  const float* rois  = (const float*)d_in[1];
  float*       out   = (float*)d_out;
  const int K = in_sizes[1] / 5;   // 1000 ROIs

  const size_t shmem =
      (size_t)(2 * FELTS           // F0, F1 (double-buffered gather)
               + 16 * 512          // V
               + 512 + 512         // Wy, Wx
               + 32 + 32           // wcol, wrow
               + 32 + 32)          // Xs, Ys
      * sizeof(float);

  roi_align_wmma<<<K, 128, shmem, stream>>>(feats, rois, out, K);
}